// PhysicsLoss_49228915146834
// MI455X (gfx1250) — compile-verified
//
#include <hip/hip_runtime.h>

typedef __attribute__((ext_vector_type(2))) float v2f;
typedef __attribute__((ext_vector_type(8))) float v8f;

#define TPB 256

// ---------------------------------------------------------------- utilities

__device__ __forceinline__ float wave_reduce_add(float v) {
  // full 32-lane (wave32) reduction
  v += __shfl_xor(v, 16, 32);
  v += __shfl_xor(v, 8, 32);
  v += __shfl_xor(v, 4, 32);
  v += __shfl_xor(v, 2, 32);
  v += __shfl_xor(v, 1, 32);
  return v;
}

// ---------------------------------------------------------------- kernel 0: zero accumulators

__global__ void zero_kernel(float* __restrict__ p, long n) {
  long i = (long)blockIdx.x * blockDim.x + threadIdx.x;
  if (i < n) p[i] = 0.0f;
}

// ---------------------------------------------------------------- kernel 1: per-element assembly

__global__ void elem_kernel(const float* __restrict__ pred,
                            const int* __restrict__ conn,
                            const float* __restrict__ dirs,
                            const float* __restrict__ len,
                            const float* __restrict__ pE,
                            const float* __restrict__ pA,
                            const float* __restrict__ pI,
                            float* __restrict__ Ftot,
                            float* __restrict__ Mtot,
                            float* __restrict__ partE,   // 2 floats per block
                            int n_elem) {
  int e = blockIdx.x * blockDim.x + threadIdx.x;
  float ea = 0.0f, eil = 0.0f;
  if (e < n_elem) {
    int i = conn[2 * e + 0];
    int j = conn[2 * e + 1];
    float xx = dirs[3 * e + 0], xy = dirs[3 * e + 1], xz = dirs[3 * e + 2];
    float L = len[e], E = pE[e], A = pA[e], I22 = pI[e];

    // local axes
    bool par = fabsf(xy) > 0.99f;
    float rx = 0.0f, ry = par ? 0.0f : 1.0f, rz = par ? 1.0f : 0.0f;
    float zx = xy * rz - xz * ry;
    float zy = xz * rx - xx * rz;
    float zz = xx * ry - xy * rx;
    float zn = fmaxf(sqrtf(zx * zx + zy * zy + zz * zz), 1e-8f);
    zx /= zn; zy /= zn; zz /= zn;
    float yx = zy * xz - zz * xy;
    float yy = zz * xx - zx * xz;
    float yz = zx * xy - zy * xx;
    float yn = fmaxf(sqrtf(yx * yx + yy * yy + yz * yz), 1e-8f);
    yx /= yn; yy /= yn; yz /= yn;

    // gather nodal DOFs
    const float* pi = pred + 6l * i;
    const float* pj = pred + 6l * j;
    float dux = pj[0] - pi[0], duy = pj[1] - pi[1], duz = pj[2] - pi[2];
    float tix = pi[3], tiy = pi[4], tiz = pi[5];
    float tjx = pj[3], tjy = pj[4], tjz = pj[5];

    float axial = dux * xx + duy * xy + duz * xz;
    float Nax = E * A / L * axial;
    float EI = E * I22;
    float L2 = L * L, L3 = L2 * L;
    float c12 = 12.0f * EI / L3, c6 = 6.0f * EI / L2, c1 = EI / L;

    float dwz = dux * zx + duy * zy + duz * zz;
    float tyi = tix * yx + tiy * yy + tiz * yz;
    float tyj = tjx * yx + tjy * yy + tjz * yz;
    float Vz  = c12 * dwz + c6 * (tyi + tyj);
    float Myi = c6 * dwz + c1 * (2.0f * tyi + tyj);
    float Myj = c6 * dwz + c1 * (tyi + 2.0f * tyj);

    float dwy = dux * yx + duy * yy + duz * yz;
    float tzi = tix * zx + tiy * zy + tiz * zz;
    float tzj = tjx * zx + tjy * zy + tjz * zz;
    float Vy  = c12 * dwy + c6 * (tzi + tzj);
    float Mzi = c6 * dwy + c1 * (2.0f * tzi + tzj);
    float Mzj = c6 * dwy + c1 * (tzi + 2.0f * tzj);

    float Fx = Nax * xx + Vz * zx + Vy * yx;
    float Fy = Nax * xy + Vz * zy + Vy * yy;
    float Fz = Nax * xz + Vz * zz + Vy * yz;

    float Mix = Myi * yx + Mzi * zx;
    float Miy = Myi * yy + Mzi * zy;
    float Miz = Myi * yz + Mzi * zz;
    float Mjx = Myj * yx + Mzj * zx;
    float Mjy = Myj * yy + Mzj * zy;
    float Mjz = Myj * yz + Mzj * zz;

    atomicAdd(&Ftot[3l * i + 0], Fx);
    atomicAdd(&Ftot[3l * i + 1], Fy);
    atomicAdd(&Ftot[3l * i + 2], Fz);
    atomicAdd(&Ftot[3l * j + 0], -Fx);
    atomicAdd(&Ftot[3l * j + 1], -Fy);
    atomicAdd(&Ftot[3l * j + 2], -Fz);
    atomicAdd(&Mtot[3l * i + 0], Mix);
    atomicAdd(&Mtot[3l * i + 1], Miy);
    atomicAdd(&Mtot[3l * i + 2], Miz);
    atomicAdd(&Mtot[3l * j + 0], Mjx);
    atomicAdd(&Mtot[3l * j + 1], Mjy);
    atomicAdd(&Mtot[3l * j + 2], Mjz);

    ea = E * A;
    eil = EI / L;
  }

  // block-level partial sums of E*A and E*I/L (deterministic tree)
  __shared__ float lds[16];
  int wave = threadIdx.x >> 5, lane = threadIdx.x & 31;
  float sea = wave_reduce_add(ea);
  float sil = wave_reduce_add(eil);
  if (lane == 0) { lds[wave] = sea; lds[8 + wave] = sil; }
  __syncthreads();
  if (threadIdx.x == 0) {
    float a0 = 0.0f, a1 = 0.0f;
    for (int w = 0; w < 8; ++w) { a0 += lds[w]; a1 += lds[8 + w]; }
    partE[2 * blockIdx.x + 0] = a0;
    partE[2 * blockIdx.x + 1] = a1;
  }
}

// ---------------------------------------------------------------- kernel 2: per-node residuals

__global__ void node_kernel(const float* __restrict__ Ftot,
                            const float* __restrict__ Mtot,
                            const float* __restrict__ line_load,
                            const float* __restrict__ bcd,
                            const float* __restrict__ bcr,
                            float* __restrict__ partN,   // 4 floats per block
                            int n_nodes) {
  int n = blockIdx.x * blockDim.x + threadIdx.x;
  float f2 = 0.0f, m2 = 0.0f, cd = 0.0f, cr = 0.0f;
  if (n < n_nodes) {
    float fd = (bcd[n] < 0.5f) ? 1.0f : 0.0f;
    float fr = (bcr[n] < 0.5f) ? 1.0f : 0.0f;
    float Fx = Ftot[3l * n + 0] + line_load[3l * n + 0];
    float Fy = Ftot[3l * n + 1] + line_load[3l * n + 1];
    float Fz = Ftot[3l * n + 2] + line_load[3l * n + 2];
    float Mx = Mtot[3l * n + 0];
    float My = Mtot[3l * n + 1];
    float Mz = Mtot[3l * n + 2];
    f2 = fd * (Fx * Fx + Fy * Fy + Fz * Fz);
    m2 = fr * (Mx * Mx + My * My + Mz * Mz);
    cd = fd;
    cr = fr;
  }
  __shared__ float lds[32];
  int wave = threadIdx.x >> 5, lane = threadIdx.x & 31;
  float v0 = wave_reduce_add(f2);
  float v1 = wave_reduce_add(m2);
  float v2 = wave_reduce_add(cd);
  float v3 = wave_reduce_add(cr);
  if (lane == 0) {
    lds[0 * 8 + wave] = v0;
    lds[1 * 8 + wave] = v1;
    lds[2 * 8 + wave] = v2;
    lds[3 * 8 + wave] = v3;
  }
  __syncthreads();
  if (threadIdx.x == 0) {
    float s0 = 0.0f, s1 = 0.0f, s2 = 0.0f, s3 = 0.0f;
    for (int w = 0; w < 8; ++w) {
      s0 += lds[0 * 8 + w]; s1 += lds[1 * 8 + w];
      s2 += lds[2 * 8 + w]; s3 += lds[3 * 8 + w];
    }
    partN[4 * blockIdx.x + 0] = s0;
    partN[4 * blockIdx.x + 1] = s1;
    partN[4 * blockIdx.x + 2] = s2;
    partN[4 * blockIdx.x + 3] = s3;
  }
}

// ---------------------------------------------------------------- kernel 3: WMMA-based final reduce
// Single wave32 (EXEC all-ones, required for WMMA). Each lane strided-accumulates
// partials into two registers (64 values wave-wide); one V_WMMA_F32_16X16X4_F32
// with an all-ones A matrix folds those 64 values into 16 f32 column sums
// (exact f32 math), then 4 xor-shuffles finish the reduction.

__device__ __forceinline__ float wmma_reduce_strided(const float* __restrict__ base,
                                                     int count, int stride, int off,
                                                     int lane) {
  float v = 0.0f, w = 0.0f;
  for (int t = lane; t < count; t += 64)      v += base[(long)t * stride + off];
  for (int t = lane + 32; t < count; t += 64) w += base[(long)t * stride + off];
  v2f a; a.x = 1.0f; a.y = 1.0f;     // A = all-ones 16x4
  v2f b; b.x = v;    b.y = w;        // B = 4x16 matrix of the 64 partials
  v8f c = {};
  c = __builtin_amdgcn_wmma_f32_16x16x4_f32(
      /*neg_a=*/false, a, /*neg_b=*/false, b,
      /*c_mod=*/(short)0, c, /*reuse_a=*/false, /*reuse_b=*/false);
  float s = c[0];                    // lane l holds colsum_{l%16}
  s += __shfl_xor(s, 1, 32);
  s += __shfl_xor(s, 2, 32);
  s += __shfl_xor(s, 4, 32);
  s += __shfl_xor(s, 8, 32);         // sum of all 16 colsums = total
  return s;
}

__global__ void finalize_kernel(const float* __restrict__ partE, int nPartE,
                                const float* __restrict__ partN, int nPartN,
                                float n_elem_f, float* __restrict__ out) {
  int lane = threadIdx.x;  // launched with exactly 32 threads
  float sumEA  = wmma_reduce_strided(partE, nPartE, 2, 0, lane);
  float sumEIL = wmma_reduce_strided(partE, nPartE, 2, 1, lane);
  float sumF2  = wmma_reduce_strided(partN, nPartN, 4, 0, lane);
  float sumM2  = wmma_reduce_strided(partN, nPartN, 4, 1, lane);
  float cntD   = wmma_reduce_strided(partN, nPartN, 4, 2, lane);
  float cntR   = wmma_reduce_strided(partN, nPartN, 4, 3, lane);
  if (lane == 0) {
    float F_char = fmaxf(sumEA / n_elem_f, 1.0f);
    float M_char = fmaxf(sumEIL / n_elem_f, 1.0f);
    float lossF = (sumF2 / (F_char * F_char)) / fmaxf(cntD * 3.0f, 1.0f);
    float lossM = (sumM2 / (M_char * M_char)) / fmaxf(cntR * 3.0f, 1.0f);
    out[0] = lossF + lossM;
  }
}

// ---------------------------------------------------------------- launcher

extern "C" void kernel_launch(void* const* d_in, const int* in_sizes, int n_in,
                              void* d_out, int out_size, void* d_ws, size_t ws_size,
                              hipStream_t stream) {
  const float* pred      = (const float*)d_in[0];
  const int*   conn      = (const int*)  d_in[1];
  const float* dirs      = (const float*)d_in[2];
  const float* len       = (const float*)d_in[3];
  const float* pE        = (const float*)d_in[4];
  const float* pA        = (const float*)d_in[5];
  const float* pI        = (const float*)d_in[6];
  const float* line_load = (const float*)d_in[7];
  const float* bcd       = (const float*)d_in[8];
  const float* bcr       = (const float*)d_in[9];

  int n_nodes = in_sizes[0] / 6;
  int n_elem  = in_sizes[3];

  int gridE = (n_elem + TPB - 1) / TPB;
  int gridN = (n_nodes + TPB - 1) / TPB;

  float* ws    = (float*)d_ws;
  float* Ftot  = ws;                          // 3*n_nodes
  float* Mtot  = Ftot + 3l * n_nodes;         // 3*n_nodes
  float* partE = Mtot + 3l * n_nodes;         // 2*gridE
  float* partN = partE + 2l * gridE;          // 4*gridN

  long nz = 6l * n_nodes;
  zero_kernel<<<(int)((nz + TPB - 1) / TPB), TPB, 0, stream>>>(Ftot, nz);
  elem_kernel<<<gridE, TPB, 0, stream>>>(pred, conn, dirs, len, pE, pA, pI,
                                         Ftot, Mtot, partE, n_elem);
  node_kernel<<<gridN, TPB, 0, stream>>>(Ftot, Mtot, line_load, bcd, bcr,
                                         partN, n_nodes);
  finalize_kernel<<<1, 32, 0, stream>>>(partE, gridE, partN, gridN,
                                        (float)n_elem, (float*)d_out);
}